// Grouper_By_NumGroup_57148834841097
// MI455X (gfx1250) — compile-verified
//
#include <hip/hip_runtime.h>
#include <hip/hip_bf16.h>
#include <math.h>

typedef float v2f __attribute__((ext_vector_type(2)));
typedef float v8f __attribute__((ext_vector_type(8)));

#define N_PTS     32768
#define FDIM      32
#define NUM_GROUP 1024
#define GROUP_SZ  32
#define RAD5      0.087266f

#define FPS_THREADS 1024
#define PPT (N_PTS / FPS_THREADS)   // 32 points per thread, held in registers

#define TILE_M   16                 // samples per block tile (one WMMA M)
#define CHUNK    512                // points per LDS slab
#define ROWPAD   516                // row stride (words) -> conflict-free stores
#define KNN_THREADS 512             // 16 waves (wave32)

// ---------------------------------------------------------------------------
// Kernel 1: farthest point sampling. One workgroup (32 waves) per batch.
// Coords + running min-distance live in registers; block argmax via LDS.
// ---------------------------------------------------------------------------
__global__ __launch_bounds__(FPS_THREADS)
void fps_kernel(const float* __restrict__ coord, int* __restrict__ s_idx) {
  const int b = blockIdx.x;
  const int t = threadIdx.x;
  const float* cb = coord + (size_t)b * N_PTS * 3;

  __shared__ float redV[32];
  __shared__ int   redI[32];
  __shared__ int   bcastIdx;

  float cx[PPT], cy[PPT], cz[PPT], d[PPT];
  const float x0 = cb[0], y0 = cb[1], z0 = cb[2];
#pragma unroll
  for (int i = 0; i < PPT; ++i) {
    const int p = i * FPS_THREADS + t;
    cx[i] = cb[p * 3 + 0]; cy[i] = cb[p * 3 + 1]; cz[i] = cb[p * 3 + 2];
    const float dx = cx[i] - x0, dy = cy[i] - y0, dz = cz[i] - z0;
    d[i] = dx * dx + dy * dy + dz * dz;
  }
  if (t == 0) s_idx[b * NUM_GROUP] = 0;

  const int lane = t & 31, wave = t >> 5;
  for (int step = 1; step < NUM_GROUP; ++step) {
    // local argmax (tie -> lowest index, matching jnp.argmax)
    float bv = -1.0f; int bi = 0x7fffffff;
#pragma unroll
    for (int i = 0; i < PPT; ++i) {
      const int p = i * FPS_THREADS + t;
      if (d[i] > bv || (d[i] == bv && p < bi)) { bv = d[i]; bi = p; }
    }
#pragma unroll
    for (int off = 16; off > 0; off >>= 1) {
      const float ov = __shfl_xor(bv, off, 32);
      const int   oi = __shfl_xor(bi, off, 32);
      if (ov > bv || (ov == bv && oi < bi)) { bv = ov; bi = oi; }
    }
    if (lane == 0) { redV[wave] = bv; redI[wave] = bi; }
    __syncthreads();
    if (wave == 0) {
      float v = redV[lane]; int ix = redI[lane];
#pragma unroll
      for (int off = 16; off > 0; off >>= 1) {
        const float ov = __shfl_xor(v, off, 32);
        const int   oi = __shfl_xor(ix, off, 32);
        if (ov > v || (ov == v && oi < ix)) { v = ov; ix = oi; }
      }
      if (lane == 0) { bcastIdx = ix; s_idx[b * NUM_GROUP + step] = ix; }
    }
    __syncthreads();
    const int fi = bcastIdx;
    const float fx = cb[fi * 3 + 0], fy = cb[fi * 3 + 1], fz = cb[fi * 3 + 2];
#pragma unroll
    for (int i = 0; i < PPT; ++i) {
      const float dx = cx[i] - fx, dy = cy[i] - fy, dz = cz[i] - fz;
      d[i] = fminf(d[i], dx * dx + dy * dy + dz * dz);
    }
    __syncthreads();   // protect bcastIdx for next iteration
  }
}

// ---------------------------------------------------------------------------
// Wave-wide bitonic compare-exchange helper (wave32).
// ---------------------------------------------------------------------------
__device__ __forceinline__
void cmpx(float& v, int& i, int j, unsigned lane, bool dirUp) {
  const float ov = __shfl_xor(v, j, 32);
  const int   oi = __shfl_xor(i, j, 32);
  const bool lower = (lane & (unsigned)j) == 0u;
  const bool otherSmaller = (ov < v) || (ov == v && oi < i);
  const bool takeOther = (dirUp == lower) ? otherSmaller : !otherSmaller;
  if (takeOther) { v = ov; i = oi; }
}

// ---------------------------------------------------------------------------
// Kernel 2: WMMA-based kNN + curvature-angle feature build.
// Block = (tile of 16 samples, batch). 16 waves cooperatively compute a
// 16x512 squared-distance slab via v_wmma_f32_16x16x4_f32 into LDS, then each
// wave maintains a bitonic top-32 for one sample row.
// ---------------------------------------------------------------------------
__global__ __launch_bounds__(KNN_THREADS)
void knn_wmma_kernel(const float* __restrict__ coord,
                     const float* __restrict__ feat,
                     const int*   __restrict__ labels,
                     const int*   __restrict__ s_idx,
                     float* __restrict__ out) {
  const int b    = blockIdx.y;
  const int tile = blockIdx.x;
  const float* cb = coord + (size_t)b * N_PTS * 3;
  const float* fb = feat  + (size_t)b * N_PTS * FDIM;

  __shared__ float sc[TILE_M][3];
  __shared__ float snorm[TILE_M];
  __shared__ int   sidx_l[TILE_M];
  __shared__ float sfeat[TILE_M][FDIM];
  __shared__ float d2t[TILE_M * ROWPAD];

  const int t    = threadIdx.x;
  const int lane = t & 31;
  const int wave = t >> 5;           // 0..15 : also sample-in-tile for selection
  const int half = lane >> 4;        // 0: K=0,1 lanes | 1: K=2,3 lanes
  const int mn   = lane & 15;        // M (for A) / N (for B) index

  if (t < TILE_M) {
    const int si = s_idx[b * NUM_GROUP + tile * TILE_M + t];
    sidx_l[t] = si;
    const float x = cb[si * 3 + 0], y = cb[si * 3 + 1], z = cb[si * 3 + 2];
    sc[t][0] = x; sc[t][1] = y; sc[t][2] = z;
    snorm[t] = x * x + y * y + z * z;
  }
  {
    // sample feature tile: thread t loads sample (t>>5), dim (t&31)
    const int s = t >> 5, dd = t & 31;
    const int si = s_idx[b * NUM_GROUP + tile * TILE_M + s];
    sfeat[s][dd] = fb[(size_t)si * FDIM + dd];
  }
  __syncthreads();

  // A fragment (16x4 f32): lanes 0-15 hold (x,y) of sample m; 16-31 hold (z,0)
  v2f afrag;
  afrag[0] = half ? sc[mn][2] : sc[mn][0];
  afrag[1] = half ? 0.0f      : sc[mn][1];
  float snr[8];
#pragma unroll
  for (int i = 0; i < 8; ++i) snr[i] = snorm[i + half * 8];

  // running top-32 per wave (sorted ascending across lanes)
  float tv = 3.402823466e38f;
  int   ti = 0x7fffffff;

  for (int c = 0; c < N_PTS / CHUNK; ++c) {
    // ---- compute phase: each wave fills 2 sub-chunks of 16 points ----
#pragma unroll
    for (int s2 = 0; s2 < 2; ++s2) {
      const int sub = wave * 2 + s2;                 // 0..31
      const int p   = c * CHUNK + sub * 16 + mn;     // global point index
      if (p + CHUNK < N_PTS) __builtin_prefetch(&cb[(p + CHUNK) * 3], 0, 1);
      const float x = cb[p * 3 + 0], y = cb[p * 3 + 1], z = cb[p * 3 + 2];
      v2f bfrag;                                     // B (4x16): mirrors A layout
      bfrag[0] = half ? z    : x;
      bfrag[1] = half ? 0.0f : y;
      const float pn = x * x + y * y + z * z;
      v8f acc = {};
      acc = __builtin_amdgcn_wmma_f32_16x16x4_f32(
          false, afrag, false, bfrag, (short)0, acc, false, false);
#pragma unroll
      for (int i = 0; i < 8; ++i) {
        const int row = i + half * 8;
        d2t[row * ROWPAD + sub * 16 + mn] = fmaf(-2.0f, acc[i], snr[i] + pn);
      }
    }
    __syncthreads();

    // ---- selection phase: wave w owns sample row w ----
#pragma unroll 1
    for (int g = 0; g < CHUNK / 32; ++g) {
      float cv = d2t[wave * ROWPAD + g * 32 + lane];
      int   ci = c * CHUNK + g * 32 + lane;
      const float thr = __shfl(tv, 31, 32);          // current 32nd-best
      if (__ballot(cv < thr) == 0ull) continue;      // wave-uniform reject
      // full bitonic sort ascending of candidates
#pragma unroll
      for (int k = 2; k <= 32; k <<= 1)
#pragma unroll
        for (int j = k >> 1; j > 0; j >>= 1)
          cmpx(cv, ci, j, (unsigned)lane, (lane & k) == 0);
      // reverse -> descending, elementwise-min against sorted top list
      const float rv = __shfl_xor(cv, 31, 32);
      const int   ri = __shfl_xor(ci, 31, 32);
      if (rv < tv || (rv == tv && ri < ti)) { tv = rv; ti = ri; }
      // result is bitonic: 5-stage merge back to ascending
#pragma unroll
      for (int j = 16; j > 0; j >>= 1)
        cmpx(tv, ti, j, (unsigned)lane, true);
    }
    __syncthreads();
  }

  // ---- epilogue: curvature angle over the 32 neighbors (lane = neighbor) ----
  float nb[FDIM];
  const float* fr = fb + (size_t)ti * FDIM;
#pragma unroll
  for (int dd = 0; dd < FDIM; ++dd) nb[dd] = fr[dd];
  float dot = 0.0f, an2 = 0.0f, bn2 = 0.0f;
#pragma unroll
  for (int dd = 0; dd < FDIM; ++dd) {
    const float a = sfeat[wave][dd];
    dot = fmaf(a, nb[dd], dot);
    an2 = fmaf(a, a, an2);
    bn2 = fmaf(nb[dd], nb[dd], bn2);
  }
  const float anbn = sqrtf(an2 * bn2);
  // ||a*bn - an*b||^2 = 2*an*bn*(anbn - dot); common factor cancels in atan2
  const float num = sqrtf(fmaxf(anbn - dot, 0.0f));
  const float den = sqrtf(fmaxf(anbn + dot, 0.0f));
  float ang = 2.0f * atan2f(num, den);   // self neighbor (lane 0) yields 0
#pragma unroll
  for (int off = 16; off > 0; off >>= 1) ang += __shfl_xor(ang, off, 32);
  const float p_curv = ang / 31.0f;      // mean over neighbors 1..31

  // ---- outputs (concatenated flat, all as float) ----
  const size_t TOT = (size_t)gridDim.y * NUM_GROUP;
  float* o_xyz  = out;
  float* o_feat = out + TOT * 3;
  float* o_sw   = o_feat + TOT * (FDIM + 1);
  float* o_lab  = o_sw + TOT;
  float* o_idx  = o_lab + TOT;
  const int r = b * NUM_GROUP + tile * TILE_M + wave;

  o_feat[(size_t)r * (FDIM + 1) + lane] = sfeat[wave][lane];
  if (lane == 0) {
    o_feat[(size_t)r * (FDIM + 1) + FDIM] = p_curv;
    o_sw[r]  = (p_curv > RAD5) ? 1.0f : 0.0f;
    o_xyz[(size_t)r * 3 + 0] = sc[wave][0];
    o_xyz[(size_t)r * 3 + 1] = sc[wave][1];
    o_xyz[(size_t)r * 3 + 2] = sc[wave][2];
    const int si = sidx_l[wave];
    o_lab[r] = (float)labels[(size_t)b * N_PTS + si];
    o_idx[r] = (float)(b * N_PTS + si);
  }
}

// ---------------------------------------------------------------------------
extern "C" void kernel_launch(void* const* d_in, const int* in_sizes, int n_in,
                              void* d_out, int out_size, void* d_ws, size_t ws_size,
                              hipStream_t stream) {
  (void)n_in; (void)out_size; (void)ws_size;
  const float* coord  = (const float*)d_in[0];
  const float* feat   = (const float*)d_in[1];
  const int*   labels = (const int*)d_in[2];
  const int b = in_sizes[3];                 // offset has one entry per batch

  int* sidx_ws = (int*)d_ws;                 // B * NUM_GROUP ints

  fps_kernel<<<b, FPS_THREADS, 0, stream>>>(coord, sidx_ws);

  dim3 grid(NUM_GROUP / TILE_M, b);
  knn_wmma_kernel<<<grid, KNN_THREADS, 0, stream>>>(
      coord, feat, labels, sidx_ws, (float*)d_out);
}